// Capsule_45475113730105
// MI455X (gfx1250) — compile-verified
//
#include <hip/hip_runtime.h>

// Capsule dynamic routing for MI455X (gfx1250).
// - u_hat recomputed per routing round via V_WMMA_F32_16X16X4_F32
//   (3 x 256MB W stream ~= 33us HBM; b/v/partials stay L2-resident).
// - W[i] blocks staged into LDS with double-buffered
//   GLOBAL_LOAD_ASYNC_TO_LDS_B128 (ASYNCcnt DMA): coalesced 512B/wave,
//   next block's DMA overlaps current block's WMMAs.
// - Batch split 2-ways across WGs (16 b each) so per-lane live set
//   (uh 32 + s_acc 32 VGPRs) fits the 128-VGPR budget of 8-waves/SIMD
//   without scratch spills; paired blocks share W[i] lines in L2.
// - Cross-lane reductions via DS_SWIZZLE_B32 XOR patterns.
// - s reduced via deterministic two-stage partial-sum (no atomics):
//   512 x 128KB partials -> fused reduce+squash kernel.

#define EPS      1e-7f
#define BATCH    32
#define N_IN     2048
#define D_IN     16
#define CAPS_N   64
#define CAPS_DIM 32
#define WG_I     8                                   // i-columns per workgroup
#define BH       16                                  // batches per workgroup
#define WBLK     (CAPS_N * CAPS_DIM * D_IN)          // floats per W[i] block (32768)
#define WBLK_B   (WBLK * 4)                          // bytes (131072)
#define PART_SZ  (BH * CAPS_N * CAPS_DIM)            // floats per WG partial (32768)
#define N_IBLK   (N_IN / WG_I)                       // 256
#define N_BLOCKS (N_IBLK * 2)                        // 512

typedef __attribute__((ext_vector_type(2))) float v2f;
typedef __attribute__((ext_vector_type(8))) float v8f;

static __device__ __forceinline__ v8f wmma_f32(v2f a, v2f b, v8f c) {
  // D = A(16x4) x B(4x16) + C(16x16), fp32
  return __builtin_amdgcn_wmma_f32_16x16x4_f32(
      /*neg_a=*/false, a, /*neg_b=*/false, b,
      /*c_mod=*/(short)0, c, /*reuse_a=*/false, /*reuse_b=*/false);
}

// ds_swizzle_b32 group-of-32 XOR add: imm = xor_mask<<10 | and_mask(0x1f)
static __device__ __forceinline__ float swz_add(float t, const int imm_xor) {
  switch (imm_xor) {  // imm must be a literal constant per ds_swizzle encoding
    case 1:  t += __int_as_float(__builtin_amdgcn_ds_swizzle(__float_as_int(t), 0x041F)); break;
    case 2:  t += __int_as_float(__builtin_amdgcn_ds_swizzle(__float_as_int(t), 0x081F)); break;
    case 4:  t += __int_as_float(__builtin_amdgcn_ds_swizzle(__float_as_int(t), 0x101F)); break;
    case 8:  t += __int_as_float(__builtin_amdgcn_ds_swizzle(__float_as_int(t), 0x201F)); break;
    case 16: t += __int_as_float(__builtin_amdgcn_ds_swizzle(__float_as_int(t), 0x401F)); break;
  }
  return t;
}

// sum across the 16 lanes of each half-wave (SWAPX1/2/4/8)
static __device__ __forceinline__ float red16_add(float t) {
  t = swz_add(t, 1); t = swz_add(t, 2); t = swz_add(t, 4); t = swz_add(t, 8);
  return t;
}

// sum across all 32 lanes (adds SWAPX16)
static __device__ __forceinline__ float red32_add(float t) {
  t = red16_add(t); t = swz_add(t, 16);
  return t;
}

// DMA one 128KB W[i] block into LDS: each of 1024 threads moves 8 x 16B.
// The same IOFFSET is applied to both global and LDS addresses
// (CDNA5 ISA), so one (lds_off, voff) pair serves all chunks.
static __device__ __forceinline__ void async_copy_w_block(
    const float* __restrict__ Wbase, unsigned gvoff, unsigned ldsoff)
{
  asm volatile(
      "global_load_async_to_lds_b128 %0, %1, %2\n\t"
      "global_load_async_to_lds_b128 %0, %1, %2 offset:16384\n\t"
      "global_load_async_to_lds_b128 %0, %1, %2 offset:32768\n\t"
      "global_load_async_to_lds_b128 %0, %1, %2 offset:49152\n\t"
      "global_load_async_to_lds_b128 %0, %1, %2 offset:65536\n\t"
      "global_load_async_to_lds_b128 %0, %1, %2 offset:81920\n\t"
      "global_load_async_to_lds_b128 %0, %1, %2 offset:98304\n\t"
      "global_load_async_to_lds_b128 %0, %1, %2 offset:114688\n\t"
      :
      : "v"(ldsoff), "v"(gvoff), "s"(Wbase)
      : "memory");
}

// One routing round. grid = 512 blocks (256 i-blocks x 2 batch-halves),
// block = 1024 threads (32 waves). Wave w owns capsules k = 2w, 2w+1 and
// computes u_hat[b0:b0+16, i, k, 0:32] = x[b0:b0+16, i, :] @ W[i,k,:,:]^T
// as 1x2 WMMA tiles, then agreement / softmax / weighted-sum in place.
__global__ __launch_bounds__(1024) void caps_round_kernel(
    const float* __restrict__ x, const float* __restrict__ W,
    const float* __restrict__ v_in, float* __restrict__ blog,
    float* __restrict__ part, const int round)
{
  extern __shared__ __align__(16) float lds_w[];   // 2 x 32768 floats (256KB)
  __shared__ float lds_x[BH * D_IN];               // x[b0:b0+16, i, :] (1 KB)
  __shared__ float lds_agr[BH][CAPS_N];            // agreement         (4 KB)
  __shared__ float lds_c[BH][CAPS_N];              // coupling coeffs   (4 KB)

  const int tid   = threadIdx.x;
  const int wave  = tid >> 5;        // 0..31
  const int lane  = tid & 31;
  const int half  = lane >> 4;       // 0 | 1 (16-lane half)
  const int l16   = lane & 15;
  const int ib    = blockIdx.x >> 1;
  const int bhalf = blockIdx.x & 1;  // which 16 batches
  const int i0    = ib * WG_I;
  const int b0    = bhalf * BH;

  const unsigned ldsw_base = (unsigned)(uintptr_t)(void*)lds_w;

  // per-wave partial s accumulators: [k_local][ntile] 16x16 tiles (32 VGPRs)
  v8f s_acc[2][2];
#pragma unroll
  for (int kl = 0; kl < 2; ++kl)
#pragma unroll
    for (int nt = 0; nt < 2; ++nt)
      s_acc[kl][nt] = (v8f){0.f,0.f,0.f,0.f,0.f,0.f,0.f,0.f};

  // prologue: DMA W[i0] into buffer 0
  async_copy_w_block(W, (unsigned)(i0 * WBLK_B) + tid * 16, ldsw_base + tid * 16);

  for (int ii = 0; ii < WG_I; ++ii) {
    const int i = i0 + ii;
    const float* wbuf = lds_w + (size_t)(ii & 1) * WBLK;

    // prefetch next W block into the other buffer (overlaps this block's WMMAs)
    if (ii + 1 < WG_I)
      async_copy_w_block(W, (unsigned)((i + 1) * WBLK_B) + tid * 16,
                         ldsw_base + (unsigned)((ii + 1) & 1) * WBLK_B + tid * 16);

    // stage x[b0:b0+16, i, :] (256 floats)
    if (tid < BH * D_IN) {
      const int bb = tid >> 4, ll = tid & 15;
      lds_x[tid] = x[(size_t)(b0 + bb) * (N_IN * D_IN) + i * D_IN + ll];
    }

    // current buffer's 8 async loads done (8 may remain in flight for prefetch)
    if (ii + 1 < WG_I) asm volatile("s_wait_asynccnt 0x8" ::: "memory");
    else               asm volatile("s_wait_asynccnt 0x0" ::: "memory");
    __syncthreads();

    // A operands (16x4 f32 layout): lane holds K = 2*half + {0,1} (+4*ks), M = l16
    v2f a_op[4];
#pragma unroll
    for (int ks = 0; ks < 4; ++ks)
      a_op[ks] = *(const v2f*)&lds_x[l16 * D_IN + ks * 4 + half * 2];

    // u_hat tiles via WMMA fp32: per (k_local, ntile): 4 K-steps
    v8f uh[2][2];
#pragma unroll
    for (int kl = 0; kl < 2; ++kl) {
      const int k = wave * 2 + kl;
#pragma unroll
      for (int nt = 0; nt < 2; ++nt) {
        const int d = nt * 16 + l16;                         // B: N = l16
        const float* Wd = wbuf + (k * CAPS_DIM + d) * D_IN + half * 2;
        v8f acc = (v8f){0.f,0.f,0.f,0.f,0.f,0.f,0.f,0.f};
#pragma unroll
        for (int ks = 0; ks < 4; ++ks)
          acc = wmma_f32(a_op[ks], *(const v2f*)&Wd[ks * 4], acc);
        uh[kl][nt] = acc;
      }
    }

    // agreement[b,i,k] = sum_d u_hat[b,i,k,d] * v[b,k,d]  (wave-local per k).
    // 32-bit offsets from the uniform v_in base -> saddr+voffset addressing,
    // no 64-bit per-lane vaddr to spill.
    if (round > 0) {
#pragma unroll
      for (int kl = 0; kl < 2; ++kl) {
        const int k = wave * 2 + kl;
#pragma unroll
        for (int r = 0; r < 8; ++r) {
          const int bl = half * 8 + r;                       // C layout: M = r + 8*half
          const unsigned vo = (unsigned)(b0 + bl) * (CAPS_N * CAPS_DIM)
                            + (unsigned)(k * CAPS_DIM) + (unsigned)l16;
          float t = uh[kl][0][r] * v_in[vo]
                  + uh[kl][1][r] * v_in[vo + 16];
          t = red16_add(t);                                  // reduce over d (16 lanes)
          if (l16 == 0) lds_agr[bl][k] = t;
        }
      }
    }
    __syncthreads();

    // softmax over k per batch (one thread per batch row); persist b-logits
    if (tid < BH) {
      const int bl = tid;
      float* crow = &lds_c[bl][0];
      float* brow = blog + ((size_t)(b0 + bl) * N_IN + i) * CAPS_N;
      float mx = -1e30f;
#pragma unroll
      for (int k = 0; k < CAPS_N; ++k) {
        float lg = 0.f;
        if (round > 0) {
          lg = lds_agr[bl][k];
          if (round == 2) lg += brow[k];   // b = b_prev + agreement
          else            brow[k] = lg;    // round 1: b_prev == 0, store
        }
        crow[k] = lg;
        mx = fmaxf(mx, lg);
      }
      float den = 0.f;
#pragma unroll
      for (int k = 0; k < CAPS_N; ++k) {
        const float e = __expf(crow[k] - mx);
        crow[k] = e;
        den += e;
      }
      const float inv = 1.f / den;
#pragma unroll
      for (int k = 0; k < CAPS_N; ++k) crow[k] *= inv;
    }
    __syncthreads();

    // s_partial[b,k,d] += c[b,i,k] * u_hat[b,i,k,d]  (register accumulate)
#pragma unroll
    for (int kl = 0; kl < 2; ++kl) {
      const int k = wave * 2 + kl;
#pragma unroll
      for (int r = 0; r < 8; ++r) {
        const int bl = half * 8 + r;
        const float c = lds_c[bl][k];      // LDS broadcast within half
#pragma unroll
        for (int nt = 0; nt < 2; ++nt)
          s_acc[kl][nt][r] += c * uh[kl][nt][r];
      }
    }
    // no trailing barrier needed: next iteration's top-of-loop wait+barrier
    // orders lds_c reads (here) before the next softmax rewrites lds_c.
  }

  // deterministic flush: plain stores into this WG's private partial slice
  float* pw = part + (size_t)blockIdx.x * PART_SZ;
#pragma unroll
  for (int kl = 0; kl < 2; ++kl) {
    const int k = wave * 2 + kl;
#pragma unroll
    for (int nt = 0; nt < 2; ++nt) {
      const int d = nt * 16 + l16;
#pragma unroll
      for (int r = 0; r < 8; ++r) {
        const int bl = half * 8 + r;
        pw[bl * (CAPS_N * CAPS_DIM) + k * CAPS_DIM + d] = s_acc[kl][nt][r];
      }
    }
  }
}

// Fused reduce + squash: one wave per (b,k) row, 8 waves per block.
// Lane = d. Sums 256 i-block partials (deterministic order), wave-reduces
// |s|^2 over d via ds_swizzle, writes v.
__global__ __launch_bounds__(256) void reduce_squash_kernel(
    const float* __restrict__ part, float* __restrict__ v_out)
{
  const int bk   = blockIdx.x * 8 + (threadIdx.x >> 5);  // 0..2047 : b*64 + k
  const int lane = threadIdx.x & 31;                     // = d
  const int b  = bk >> 6, k = bk & 63;
  const int h  = b >> 4,  bl = b & 15;  // batch-half, row within half
  const float* base = part + (size_t)h * PART_SZ
                    + bl * (CAPS_N * CAPS_DIM) + k * CAPS_DIM + lane;
  float acc = 0.f;
#pragma unroll 4
  for (int j = 0; j < N_IBLK; ++j)      // partial index = j*2 + h
    acc += base[(size_t)j * (2 * PART_SZ)];
  const float sq = red32_add(acc * acc);
  const float s2 = sq + EPS;
  const float scale = sqrtf(s2) / (1.f + s2);
  v_out[(size_t)bk * CAPS_DIM + lane] = scale * acc;
}

extern "C" void kernel_launch(void* const* d_in, const int* in_sizes, int n_in,
                              void* d_out, int out_size, void* d_ws, size_t ws_size,
                              hipStream_t stream) {
  const float* x = (const float*)d_in[0];   // [32, 2048, 16]
  const float* W = (const float*)d_in[1];   // [2048, 64, 32, 16]
  float* out = (float*)d_out;               // [32, 64, 32]

  // workspace layout (floats): partials | v | b-logits  (~81 MB)
  float* part  = (float*)d_ws;                        // 512*32768 = 16.78M floats
  float* v_buf = part + (size_t)N_BLOCKS * PART_SZ;   // 65536
  float* b_buf = v_buf + BATCH * CAPS_N * CAPS_DIM;   // 32*2048*64 = 4.19M

  const int kDynLds = 2 * WBLK_B;                     // 256KB double buffer
  const dim3 rgrid(N_BLOCKS);                         // 512
  const dim3 rblk(1024);
  const dim3 qgrid(BATCH * CAPS_N / 8);               // 256
  const dim3 qblk(256);

  // allow >48KB dynamic LDS (gfx1250 WGP has 320KB); idempotent, capture-safe
  (void)hipFuncSetAttribute((const void*)caps_round_kernel,
                            hipFuncAttributeMaxDynamicSharedMemorySize, kDynLds);

  // round 0: c uniform (softmax of zeros)
  caps_round_kernel<<<rgrid, rblk, kDynLds, stream>>>(x, W, v_buf, b_buf, part, 0);
  reduce_squash_kernel<<<qgrid, qblk, 0, stream>>>(part, v_buf);

  // round 1: b = agreement(v0); writes b_buf
  caps_round_kernel<<<rgrid, rblk, kDynLds, stream>>>(x, W, v_buf, b_buf, part, 1);
  reduce_squash_kernel<<<qgrid, qblk, 0, stream>>>(part, v_buf);

  // round 2: b += agreement(v1); final v -> d_out
  caps_round_kernel<<<rgrid, rblk, kDynLds, stream>>>(x, W, v_buf, b_buf, part, 2);
  reduce_squash_kernel<<<qgrid, qblk, 0, stream>>>(part, out);
}